// MultiSubnetLinear_85529978732853
// MI455X (gfx1250) — compile-verified
//
#include <hip/hip_runtime.h>
#include <hip/hip_bf16.h>

#define NUM_MODELS 64
#define IN_SUB     128          // IN_F / F1
#define OUT_SUB    128          // OUT_F / F0
#define BATCH      32768
#define W_ROW      256          // full IN_F stride of weight rows
#define W_MODEL    (256 * 256)  // full per-model weight stride
#define B_ROW      256          // per-model bias stride
#define MAX_TILES  (BATCH / 16 + NUM_MODELS)

typedef __attribute__((ext_vector_type(2))) float v2f;
typedef __attribute__((ext_vector_type(8))) float v8f;

struct GroupWs {
    int counts[NUM_MODELS];
    int cursor[NUM_MODELS];
    int numTiles;
    int pad_;
    int tileM[MAX_TILES];
    int tileRow[MAX_TILES];
    int tileValid[MAX_TILES];
    int perm[BATCH];
};

// ---- pass 1: zero histogram ------------------------------------------------
__global__ void msl_init_kernel(GroupWs* ws) {
    int t = threadIdx.x;
    if (t < NUM_MODELS) ws->counts[t] = 0;
}

// ---- pass 2: histogram of model indices ------------------------------------
__global__ void msl_hist_kernel(const int* __restrict__ idx, GroupWs* ws) {
    int b = blockIdx.x * blockDim.x + threadIdx.x;
    if (b < BATCH) {
        int m = idx[b] & (NUM_MODELS - 1);
        atomicAdd(&ws->counts[m], 1);
    }
}

// ---- pass 3: exclusive scan + build 16-row tile descriptors (serial, tiny) --
__global__ void msl_build_kernel(GroupWs* ws) {
    if (threadIdx.x != 0 || blockIdx.x != 0) return;
    int off = 0;
    int nt  = 0;
    for (int m = 0; m < NUM_MODELS; ++m) {
        int c = ws->counts[m];
        ws->cursor[m] = off;
        for (int t = 0; t < c; t += 16) {
            int rem = c - t;
            ws->tileM[nt]     = m;
            ws->tileRow[nt]   = off + t;
            ws->tileValid[nt] = rem < 16 ? rem : 16;
            ++nt;
        }
        off += c;
    }
    ws->numTiles = nt;
}

// ---- pass 4: scatter row ids grouped by model ------------------------------
__global__ void msl_scatter_kernel(const int* __restrict__ idx, GroupWs* ws) {
    int b = blockIdx.x * blockDim.x + threadIdx.x;
    if (b < BATCH) {
        int m   = idx[b] & (NUM_MODELS - 1);
        int pos = atomicAdd(&ws->cursor[m], 1);
        ws->perm[pos] = b;
    }
}

// ---- pass 5: grouped GEMM, one 16-row tile per block, 8 WMMA waves ----------
// out[r, o] = sum_k W[m, o, k] * x[r, k] + bias[m, o]
__global__ void __launch_bounds__(256)
msl_gemm_kernel(const float* __restrict__ x,
                const float* __restrict__ w,
                const float* __restrict__ bias,
                float* __restrict__ out,
                GroupWs* ws) {
    const int j = blockIdx.x;
    if (j >= ws->numTiles) return;

    const int m        = ws->tileM[j];
    const int rowStart = ws->tileRow[j];
    const int valid    = ws->tileValid[j];

    // Row stride 132 floats: lane L reads Xs[L&15][k + 2*(L>>4)] as b64;
    // bank = (132*r + c) % 64 = (4r + 2h + c) % 64 -> all 32 lanes hit
    // disjoint bank pairs (no conflicts).
    __shared__ float Xs[16][132];
    __shared__ int   rows[16];

    const int t = threadIdx.x;
    if (t < 16) rows[t] = (t < valid) ? ws->perm[rowStart + t] : -1;
    __syncthreads();

    // Cooperative X-tile load: 16 rows x 128 cols, 8 floats per thread.
    {
        const int r  = t >> 4;          // 0..15
        const int cb = (t & 15) * 8;    // column base
        const int grow = rows[r];
        if (grow >= 0) {
            const float4* src = (const float4*)(x + (long)grow * IN_SUB + cb);
            *(float4*)&Xs[r][cb]     = src[0];
            *(float4*)&Xs[r][cb + 4] = src[1];
        } else {
            const float4 z = {0.f, 0.f, 0.f, 0.f};
            *(float4*)&Xs[r][cb]     = z;
            *(float4*)&Xs[r][cb + 4] = z;
        }
    }
    __syncthreads();

    const int lane  = t & 31;
    const int wv    = t >> 5;       // 0..7 -> output column tile
    const int r     = lane & 15;    // A: M row / B: N col (per ISA layout)
    const int h     = lane >> 4;    // half-wave selects K pair {0,1} vs {2,3}
    const int obase = wv * 16;
    const int k2    = 2 * h;

    // W row for B-fragment: B[kk][n] = W[m, obase + n, k + kk]
    const float* wrow = w + (long)m * W_MODEL + (long)(obase + r) * W_ROW;

    v8f acc = {};
    #pragma unroll
    for (int k = 0; k < IN_SUB; k += 4) {
        v2f a;
        a.x = Xs[r][k + k2];
        a.y = Xs[r][k + k2 + 1];
        const float2 bw = *(const float2*)(wrow + k + k2);
        v2f b;
        b.x = bw.x;
        b.y = bw.y;
        acc = __builtin_amdgcn_wmma_f32_16x16x4_f32(
            /*neg_a=*/false, a, /*neg_b=*/false, b,
            /*c_mod=*/(short)0, acc, /*reuse_a=*/false, /*reuse_b=*/false);
    }

    const float bv = bias[(long)m * B_ROW + obase + r];

    // D layout: VGPR jrow -> (M = jrow + 8*h, N = lane&15)
    #pragma unroll
    for (int jrow = 0; jrow < 8; ++jrow) {
        const int Mrow = jrow + 8 * h;
        const int grow = rows[Mrow];
        if (grow >= 0) {
            out[(long)grow * OUT_SUB + obase + r] = acc[jrow] + bv;
        }
    }
}

extern "C" void kernel_launch(void* const* d_in, const int* in_sizes, int n_in,
                              void* d_out, int out_size, void* d_ws, size_t ws_size,
                              hipStream_t stream) {
    const float* x    = (const float*)d_in[0];   // (32768, 128)
    const float* w    = (const float*)d_in[1];   // (64, 256, 256)
    const float* bias = (const float*)d_in[2];   // (64, 256)
    const int*   idx  = (const int*)d_in[3];     // (32768,)
    float* out = (float*)d_out;                  // (32768, 128)
    GroupWs* ws = (GroupWs*)d_ws;

    msl_init_kernel<<<1, 64, 0, stream>>>(ws);
    msl_hist_kernel<<<BATCH / 256, 256, 0, stream>>>(idx, ws);
    msl_build_kernel<<<1, 1, 0, stream>>>(ws);
    msl_scatter_kernel<<<BATCH / 256, 256, 0, stream>>>(idx, ws);
    msl_gemm_kernel<<<MAX_TILES, 256, 0, stream>>>(x, w, bias, out, ws);
}